// CH_LSTM_77421080478506
// MI455X (gfx1250) — compile-verified
//
#include <hip/hip_runtime.h>
#include <math.h>
#include <stdint.h>

// LSTM: T=512 steps, B=64 batch, DIN=1024, D=1024, K = DIN+D = 2048.
// fp32 WMMA (V_WMMA_F32_16X16X4_F32); weights staged to LDS with
// GLOBAL_LOAD_ASYNC_TO_LDS_B128 (ASYNCcnt), double-buffered.

typedef __attribute__((ext_vector_type(2))) float v2f;
typedef __attribute__((ext_vector_type(8))) float v8f;

namespace {
constexpr int T_STEPS = 512;
constexpr int BN      = 64;       // batch (== all M rows per block)
constexpr int DIN     = 1024;
constexpr int D       = 1024;
constexpr int KTOT    = DIN + D;  // 2048
constexpr int KC      = 64;       // K chunk staged in LDS
}

__device__ __forceinline__ void async_copy_b128(uint32_t lds_off, const void* gptr) {
  // VDST = LDS byte address, GV addressing (64-bit VGPR addr, saddr = off).
  asm volatile("global_load_async_to_lds_b128 %0, %1, off"
               :: "v"(lds_off), "v"(gptr)
               : "memory");
}

__device__ __forceinline__ void wait_async_zero() {
  asm volatile("s_wait_asynccnt 0" ::: "memory");
}

__global__ __launch_bounds__(256) void lstm_init_kernel(
    const float* __restrict__ c0, float* __restrict__ c_state, float* __restrict__ h0) {
  int i = blockIdx.x * blockDim.x + threadIdx.x;
  if (i < BN * D) {
    float c = c0[i % D];
    c_state[i] = c;
    h0[i] = tanhf(c);
  }
}

__global__ __launch_bounds__(256) void lstm_step_kernel(
    const float* __restrict__ batch_t,   // [BN][DIN] slice for this t
    const float* __restrict__ h_prev,    // [BN][D]
    const float* __restrict__ Wi, const float* __restrict__ Wf,
    const float* __restrict__ Wc, const float* __restrict__ Wo,
    const float* __restrict__ bi, const float* __restrict__ bf,
    const float* __restrict__ bc, const float* __restrict__ bo,
    const int*   __restrict__ lengths,
    float* __restrict__ c_state,         // [BN][D]
    float* __restrict__ out_t,           // [BN][D] slice of d_out for this t
    int t) {
  const int tid  = threadIdx.x;
  const int lane = tid & 31;       // wave32
  const int wave = tid >> 5;       // 0..7
  const int g    = wave & 3;       // gate id (0=i,1=f,2=c,3=o)
  const int p    = wave >> 2;      // m-half: rows [32p, 32p+32)
  const int half = lane >> 4;      // K-half select for f32 fragments
  const int idx  = lane & 15;      // M (A) / N (B) index within tile

  const int n_base = blockIdx.x * 16;   // 0..1008
  const int mb0 = p * 32;               // first 16x16 m-tile base
  const int mb1 = p * 32 + 16;          // second m-tile base

  __shared__ float sA[2][KC * 64];      // [buf][k][m] : shared x panel (all 64 rows)
  __shared__ float sB[2][4][KC * 16];   // [buf][gate][k][n]

  const float* Wg = (g == 0) ? Wi : (g == 1) ? Wf : (g == 2) ? Wc : Wo;

  v8f acc0 = {0.f, 0.f, 0.f, 0.f, 0.f, 0.f, 0.f, 0.f};
  v8f acc1 = {0.f, 0.f, 0.f, 0.f, 0.f, 0.f, 0.f, 0.f};

  // ---- A staging mapping: thread -> (row, 16-col slab) ----
  const int arow = tid >> 2;            // 0..63
  const int acol = (tid & 3) * 16;      // 0,16,32,48
  // ---- B staging mapping: gate g staged by waves g (rows 0..31) and g+4 (32..63)
  const int brow0 = p * 32 + (lane >> 2);  // + pass*8
  const int bnn   = (lane & 3) * 4;

  auto stageA = [&](int buf, int k0) {
    const int kk = k0 + acol;  // KC chunks never straddle the DIN boundary
    const float* src = (k0 < DIN)
        ? (batch_t + (size_t)arow * DIN + kk)
        : (h_prev  + (size_t)arow * D   + (kk - DIN));
    float4 p0 = *(const float4*)(src);
    float4 p1 = *(const float4*)(src + 4);
    float4 p2 = *(const float4*)(src + 8);
    float4 p3 = *(const float4*)(src + 12);
    float v[16] = {p0.x,p0.y,p0.z,p0.w, p1.x,p1.y,p1.z,p1.w,
                   p2.x,p2.y,p2.z,p2.w, p3.x,p3.y,p3.z,p3.w};
#pragma unroll
    for (int j = 0; j < 16; ++j) {
      sA[buf][(acol + j) * 64 + arow] = v[j];
    }
  };

  auto stageB = [&](int buf, int k0) {
#pragma unroll
    for (int pass = 0; pass < 4; ++pass) {
      const int r = brow0 + pass * 8;
      const uint32_t loff = (uint32_t)(size_t)&sB[buf][g][r * 16 + bnn];
      const float* gp = Wg + (size_t)(k0 + r) * D + n_base + bnn;
      async_copy_b128(loff, gp);
    }
  };

  auto compute = [&](int buf) {
#pragma unroll
    for (int kk = 0; kk < KC; kk += 4) {
      const int ka = kk + 2 * half;  // lanes 16-31 hold K+2 (ISA f32 layout)
      v2f b, a0, a1;
      b.x  = sB[buf][g][ka * 16 + idx];
      b.y  = sB[buf][g][(ka + 1) * 16 + idx];
      a0.x = sA[buf][ka * 64 + mb0 + idx];
      a0.y = sA[buf][(ka + 1) * 64 + mb0 + idx];
      a1.x = sA[buf][ka * 64 + mb1 + idx];
      a1.y = sA[buf][(ka + 1) * 64 + mb1 + idx];
      acc0 = __builtin_amdgcn_wmma_f32_16x16x4_f32(
          false, a0, false, b, (short)0, acc0, false, false);
      acc1 = __builtin_amdgcn_wmma_f32_16x16x4_f32(
          false, a1, false, b, (short)0, acc1, false, false);
    }
  };

  // ---- pipelined main loop (double-buffered LDS) ----
  stageB(0, 0);
  stageA(0, 0);
  wait_async_zero();
  __syncthreads();

  int bufc = 0;
  for (int k0 = KC; k0 < KTOT; k0 += KC) {
    stageB(bufc ^ 1, k0);   // async weight panel for chunk i
    stageA(bufc ^ 1, k0);   // activation panel for chunk i
    compute(bufc);          // WMMAs on chunk i-1
    wait_async_zero();
    __syncthreads();
    bufc ^= 1;
  }
  compute(bufc);

  // ---- exchange gate tiles through LDS (C layout: M = v + 8*half, N = idx) ----
  float* sR = sA[0];  // 4096 floats, free after the last-but-one barrier: [g][m][n]
#pragma unroll
  for (int v = 0; v < 8; ++v) {
    const int m0 = mb0 + v + 8 * half;
    sR[(g * 64 + m0) * 16 + idx]        = acc0[v];
    sR[(g * 64 + m0 + 16) * 16 + idx]   = acc1[v];
  }
  __syncthreads();

  // ---- fused cell update: 64x16 elements, 4 per thread ----
#pragma unroll
  for (int e = tid; e < BN * 16; e += 256) {
    const int m   = e >> 4;        // batch row
    const int n   = e & 15;
    const int col = n_base + n;
    const float pi = sR[(0 * 64 + m) * 16 + n] + bi[col];
    const float pf = sR[(1 * 64 + m) * 16 + n] + bf[col];
    const float pc = sR[(2 * 64 + m) * 16 + n] + bc[col];
    const float po = sR[(3 * 64 + m) * 16 + n] + bo[col];
    const float ig = 1.f / (1.f + expf(-pi));
    const float fg = 1.f / (1.f + expf(-pf));
    const float gg = tanhf(pc);
    const float og = 1.f / (1.f + expf(-po));
    const size_t off = (size_t)m * D + col;
    const float cn = fg * c_state[off] + ig * gg;
    c_state[off] = cn;
    float h = og * tanhf(cn);
    if (t >= lengths[m]) h = 0.f;  // masked h is emitted AND carried (via d_out)
    out_t[off] = h;
  }
}

extern "C" void kernel_launch(void* const* d_in, const int* in_sizes, int n_in,
                              void* d_out, int out_size, void* d_ws, size_t ws_size,
                              hipStream_t stream) {
  const float* batch   = (const float*)d_in[0];
  const int*   lengths = (const int*)d_in[1];
  const float* c0      = (const float*)d_in[2];
  const float* Wi      = (const float*)d_in[3];
  const float* bi      = (const float*)d_in[4];
  const float* Wf      = (const float*)d_in[5];
  const float* bf      = (const float*)d_in[6];
  const float* Wc      = (const float*)d_in[7];
  const float* bc      = (const float*)d_in[8];
  const float* Wo      = (const float*)d_in[9];
  const float* bo      = (const float*)d_in[10];
  float* out = (float*)d_out;

  float* c_state = (float*)d_ws;           // [BN*D] floats (256 KB)
  float* h0      = c_state + BN * D;       // [BN*D] floats (256 KB)

  lstm_init_kernel<<<(BN * D + 255) / 256, 256, 0, stream>>>(c0, c_state, h0);

  for (int t = 0; t < T_STEPS; ++t) {
    const float* h_prev = (t == 0) ? h0 : (out + (size_t)(t - 1) * BN * D);
    lstm_step_kernel<<<dim3(64), 256, 0, stream>>>(
        batch + (size_t)t * BN * DIN, h_prev,
        Wi, Wf, Wc, Wo, bi, bf, bc, bo,
        lengths, c_state, out + (size_t)t * BN * D, t);
  }
}